// FocalCTLoss_31198642438365
// MI455X (gfx1250) — compile-verified
//
#include <hip/hip_runtime.h>

// Problem dimensions (fixed by the reference setup).
#define T_DIM 512
#define B_DIM 256
#define C_DIM 256
#define S_DIM 64
#define L_DIM 129          // 2*S + 1
#define NEGV  (-1e30f)
#define DEPTH 8            // async row-pipeline depth (ring of 1KB row buffers)

#define AS1 __attribute__((address_space(1)))
#define AS3 __attribute__((address_space(3)))

#if defined(__gfx1250__) && __has_builtin(__builtin_amdgcn_global_load_async_to_lds_b32) && __has_builtin(__builtin_amdgcn_s_wait_asynccnt)
#define USE_ASYNC 1
#else
#define USE_ASYNC 0
#endif

// Address-space cast helpers (builtin wants non-const AS1/AS3 int* pointers).
__device__ __forceinline__ AS1 int* as_global(const float* p) {
    return (AS1 int*)(float*)p;          // strip const, then addrspacecast
}
__device__ __forceinline__ AS3 int* as_lds(float* p) {
    return (AS3 int*)p;
}

__device__ __forceinline__ float lae2(float a, float b) {
    float m = fmaxf(a, b);
    return m + __logf(__expf(a - m) + __expf(b - m));
}

__device__ __forceinline__ float lae3(float a, float b, float c) {
    float m = fmaxf(a, fmaxf(b, c));
    return m + __logf(__expf(a - m) + __expf(b - m) + __expf(c - m));
}

// One workgroup per batch element. 256 threads = 8 wave32.
// Threads 0..L-1 each own one extended-label position (state in registers).
// All 256 threads cooperate: each wave async-copies a 128B slice of each
// 1KB log-prob row, so every wave issues EXACTLY ONE async op per row and
// the in-order ASYNCcnt math below is uniform across waves.
__global__ __launch_bounds__(256) void ctc_alpha_kernel(
    const float* __restrict__ log_probs,      // (T, B, C)
    const int*   __restrict__ targets,        // (B, S)
    const int*   __restrict__ input_lengths,  // (B,)
    const int*   __restrict__ target_lengths, // (B,)
    float*       __restrict__ per_batch)      // (B,) focal loss per element
{
    __shared__ float rbuf[DEPTH][C_DIM];   // ring of log_probs rows (8 x 1KB)
    __shared__ float alpha[2][L_DIM + 2];  // ping-pong alpha, 2-slot NEG pad at front

    const int b   = blockIdx.x;
    const int tid = threadIdx.x;
    const size_t tstride = (size_t)B_DIM * C_DIM;
    const float* base = log_probs + (size_t)b * C_DIM;   // row (t=0, b)

    // Per-thread DP constants: extended label + skip-transition flag.
    int  lab  = 0;
    bool skip = false;
    if (tid < L_DIM) {
        if (tid & 1) lab = targets[b * S_DIM + (tid >> 1)];
        skip = (tid >= 3) && (tid & 1) &&
               (targets[b * S_DIM + (tid >> 1)] != targets[b * S_DIM + (tid >> 1) - 1]);
    }
    const int len = input_lengths[b];

    // NEG pads (never rewritten; DP writes only index >= 2).
    if (tid < 2) { alpha[0][tid] = NEGV; alpha[1][tid] = NEGV; }

    // Fill the pipeline: issue rows 0..DEPTH-1 (one async op per wave per row).
#if USE_ASYNC
    #pragma unroll
    for (int r = 0; r < DEPTH; ++r) {
        __builtin_amdgcn_global_load_async_to_lds_b32(
            as_global(base + (size_t)r * tstride + tid),
            as_lds(&rbuf[r][tid]), 0, 0);
    }
    __builtin_amdgcn_s_wait_asynccnt(DEPTH - 1);   // in-order -> row 0 landed
#else
    #pragma unroll
    for (int r = 0; r < DEPTH; ++r)
        rbuf[r][tid] = base[(size_t)r * tstride + tid];
#endif
    __syncthreads();

    // alpha at t=0: only positions 0 (blank) and 1 (first label) are live.
    if (tid < L_DIM) {
        float a0 = NEGV;
        if (tid == 0)      a0 = rbuf[0][0];
        else if (tid == 1) a0 = rbuf[0][lab];
        alpha[0][tid + 2] = a0;
    }

    // Main scan. Step t: reads alpha[(t-1)&1], writes alpha[t&1], uses
    // rbuf[t % DEPTH]. Invariant at loop top: rows <= t+DEPTH-2 have been
    // issued; allowing DEPTH-2 outstanding ops per wave (newest = rows
    // t+1..t+DEPTH-2) forces row t complete BEFORE the barrier, so the
    // barrier certifies all waves' slices of row t are in LDS. The buffer
    // refilled below ((t-1) % DEPTH) was last read in step t-1, i.e. before
    // this barrier -> no WAR hazard.
    for (int t = 1; t < T_DIM; ++t) {
#if USE_ASYNC
        __builtin_amdgcn_s_wait_asynccnt(DEPTH - 2);
#endif
        __syncthreads();   // row t complete in LDS; alpha writes of step t-1 visible

        if (t + DEPTH - 1 < T_DIM) {
            const float* gp = base + (size_t)(t + DEPTH - 1) * tstride + tid;
#if USE_ASYNC
            __builtin_amdgcn_global_load_async_to_lds_b32(
                as_global(gp), as_lds(&rbuf[(t + DEPTH - 1) % DEPTH][tid]), 0, 0);
#else
            rbuf[(t + DEPTH - 1) % DEPTH][tid] = *gp;
#endif
        }

        if (tid < L_DIM) {
            const float* src = alpha[(t + 1) & 1];   // == (t-1)&1
            float*       dst = alpha[t & 1];
            float a0 = src[tid + 2];                  // stay
            float a1 = src[tid + 1];                  // from j-1
            float a2 = skip ? src[tid] : NEGV;        // from j-2 (if allowed)
            float emit = rbuf[t % DEPTH][lab];        // LDS gather of log_probs[t,b,ext[j]]
            float nv = lae3(a0, a1, a2) + emit;
            dst[tid + 2] = (t < len) ? nv : a0;       // freeze ended sequences
        }
    }
    __syncthreads();

    if (tid == 0) {
        const float* fin = alpha[(T_DIM - 1) & 1];
        int il = 2 * target_lengths[b];
        float l1 = fin[il + 2];
        float l2 = fin[il + 1];
        float loss = -lae2(l1, l2);
        float pt = __expf(-loss);
        float w = 1.0f - pt;
        per_batch[b] = w * w * loss;   // GAMMA = 2
    }
}

// Deterministic fixed-order tree reduction -> mean over batch.
__global__ __launch_bounds__(256) void focal_mean_kernel(
    const float* __restrict__ per_batch, float* __restrict__ out)
{
    __shared__ float s[B_DIM];
    int tid = threadIdx.x;
    s[tid] = per_batch[tid];
    __syncthreads();
    for (int off = B_DIM / 2; off > 0; off >>= 1) {
        if (tid < off) s[tid] += s[tid + off];
        __syncthreads();
    }
    if (tid == 0) out[0] = s[0] * (1.0f / (float)B_DIM);
}

extern "C" void kernel_launch(void* const* d_in, const int* in_sizes, int n_in,
                              void* d_out, int out_size, void* d_ws, size_t ws_size,
                              hipStream_t stream) {
    const float* log_probs      = (const float*)d_in[0];
    const int*   targets        = (const int*)d_in[1];
    const int*   input_lengths  = (const int*)d_in[2];
    const int*   target_lengths = (const int*)d_in[3];
    float* out = (float*)d_out;
    float* ws  = (float*)d_ws;   // B floats of per-batch focal loss

    ctc_alpha_kernel<<<B_DIM, 256, 0, stream>>>(log_probs, targets,
                                                input_lengths, target_lengths, ws);
    focal_mean_kernel<<<1, B_DIM, 0, stream>>>(ws, out);
}